// TopoGCN_59828894433563
// MI455X (gfx1250) — compile-verified
//
#include <hip/hip_runtime.h>
#include <hip/hip_bf16.h>

#define HID 512

typedef float v2f __attribute__((ext_vector_type(2)));
typedef float v8f __attribute__((ext_vector_type(8)));

// ---------------------------------------------------------------------------
// CSR construction (deterministic integer atomics)
// ---------------------------------------------------------------------------
__global__ void zero_i32(int* __restrict__ p, int n) {
    int i = blockIdx.x * blockDim.x + threadIdx.x;
    if (i < n) p[i] = 0;
}

__global__ void count_edges(const int* __restrict__ row, int* __restrict__ counts, int E) {
    int e = blockIdx.x * blockDim.x + threadIdx.x;
    if (e < E) atomicAdd(&counts[row[e]], 1);
}

// Single-block exclusive scan over N row counts -> rowptr[0..N]
__global__ __launch_bounds__(1024)
void scan_rows(const int* __restrict__ counts, int* __restrict__ rowptr, int n) {
    __shared__ int sm[1024];
    __shared__ int carry;
    if (threadIdx.x == 0) { carry = 0; rowptr[0] = 0; }
    __syncthreads();
    int nChunks = (n + 1023) >> 10;
    for (int c = 0; c < nChunks; ++c) {
        int i = (c << 10) + (int)threadIdx.x;
        int v = (i < n) ? counts[i] : 0;
        sm[threadIdx.x] = v;
        __syncthreads();
        for (int off = 1; off < 1024; off <<= 1) {
            int t = (threadIdx.x >= (unsigned)off) ? sm[threadIdx.x - off] : 0;
            __syncthreads();
            sm[threadIdx.x] += t;
            __syncthreads();
        }
        if (i < n) rowptr[i + 1] = carry + sm[threadIdx.x];
        __syncthreads();
        if (threadIdx.x == 0) carry += sm[1023];
        __syncthreads();
    }
}

__global__ void scatter_edges(const int* __restrict__ row, const int* __restrict__ col,
                              const float* __restrict__ val, const int* __restrict__ rowptr,
                              int* __restrict__ fill, int* __restrict__ scols,
                              float* __restrict__ svals, int E) {
    int e = blockIdx.x * blockDim.x + threadIdx.x;
    if (e < E) {
        int r = row[e];
        int pos = rowptr[r] + atomicAdd(&fill[r], 1);
        scols[pos] = col[e];
        svals[pos] = val[e];
    }
}

// ---------------------------------------------------------------------------
// 512x512 transpose (LDS-tiled, coalesced both ways): Bt[n][k] = B[k][n]
// Makes each WMMA B-fragment a contiguous 8-byte load in the GEMM.
// ---------------------------------------------------------------------------
__global__ __launch_bounds__(256)
void transpose512(const float* __restrict__ B, float* __restrict__ Bt) {
    __shared__ float tile[32][33];
    const int bx = blockIdx.x * 32;           // source column tile
    const int by = blockIdx.y * 32;           // source row tile
    const int tx = threadIdx.x & 31;
    const int ty = threadIdx.x >> 5;          // 0..7
#pragma unroll
    for (int i = 0; i < 32; i += 8)
        tile[ty + i][tx] = B[(size_t)(by + ty + i) * HID + bx + tx];
    __syncthreads();
#pragma unroll
    for (int i = 0; i < 32; i += 8)
        Bt[(size_t)(bx + ty + i) * HID + by + tx] = tile[tx][ty + i];
}

// ---------------------------------------------------------------------------
// Dense GEMM  C[M x 512] = A[M x 512] * B  via V_WMMA_F32_16X16X4_F32,
// with B pre-transposed (Bt[n][k]) so every fragment load is a b64.
//   grid.x = ceil(M/16) row tiles; 256 threads = 8 waves; wave w -> cols [64w,64w+64)
//   ISA 7.12.2 (32-bit 16x4 A): lane L holds A[m=L&15][kb + 2*(L>>4) .. +1]
//   B (4x16) mirrors with n = L&15;  C: VGPR j -> row base+j / base+j+8 by half-wave.
//   Out-of-range rows are CLAMPED on load (their C rows are masked at store).
// ---------------------------------------------------------------------------
__global__ __launch_bounds__(256)
void gemm_wmma_f32(const float* __restrict__ A, const float* __restrict__ Bt,
                   const float* __restrict__ bias, float* __restrict__ C,
                   int M, int applyBiasRelu) {
    const int lane  = threadIdx.x & 31;
    const int wave  = threadIdx.x >> 5;       // 0..7 -> 64-column slab
    const int mrow  = lane & 15;
    const int khalf = lane >> 4;              // which K pair this half-wave owns
    const int rowBase = blockIdx.x * 16;
    const int colBase = wave * 64;

    v8f acc0 = {}, acc1 = {}, acc2 = {}, acc3 = {};

    const int m = min(rowBase + mrow, M - 1);                  // clamp: no branch in loop
    const float* arow   = A  + (size_t)m * HID + khalf * 2;
    const float* btbase = Bt + (size_t)(colBase + mrow) * HID + khalf * 2;

    for (int k = 0; k < HID; k += 4) {
        v2f a  = *(const v2f*)(arow + k);
        v2f b0 = *(const v2f*)(btbase + k);
        v2f b1 = *(const v2f*)(btbase + 16 * HID + k);
        v2f b2 = *(const v2f*)(btbase + 32 * HID + k);
        v2f b3 = *(const v2f*)(btbase + 48 * HID + k);
        acc0 = __builtin_amdgcn_wmma_f32_16x16x4_f32(false, a, false, b0, (short)0, acc0, false, false);
        acc1 = __builtin_amdgcn_wmma_f32_16x16x4_f32(false, a, false, b1, (short)0, acc1, false, false);
        acc2 = __builtin_amdgcn_wmma_f32_16x16x4_f32(false, a, false, b2, (short)0, acc2, false, false);
        acc3 = __builtin_amdgcn_wmma_f32_16x16x4_f32(false, a, false, b3, (short)0, acc3, false, false);
    }

    const int srow0 = rowBase + khalf * 8;
    const int cbase = colBase + mrow;
#pragma unroll
    for (int j = 0; j < 8; ++j) {
        int r = srow0 + j;
        if (r < M) {
            float v0 = acc0[j], v1 = acc1[j], v2 = acc2[j], v3 = acc3[j];
            if (applyBiasRelu) {
                v0 = fmaxf(v0 + bias[cbase +  0], 0.f);
                v1 = fmaxf(v1 + bias[cbase + 16], 0.f);
                v2 = fmaxf(v2 + bias[cbase + 32], 0.f);
                v3 = fmaxf(v3 + bias[cbase + 48], 0.f);
            }
            float* crow = C + (size_t)r * HID + cbase;
            crow[ 0] = v0; crow[16] = v1; crow[32] = v2; crow[48] = v3;
        }
    }
}

// ---------------------------------------------------------------------------
// SpMM gather + bias + ReLU: out[r,:] = relu(bias + sum_e val[e]*h[col[e],:])
// One block per row; 128 threads x float4 -> 512 columns (b128 gathers).
// ---------------------------------------------------------------------------
__global__ __launch_bounds__(128)
void spmm_bias_relu(const int* __restrict__ rowptr, const int* __restrict__ cols,
                    const float* __restrict__ vals, const float* __restrict__ h,
                    const float* __restrict__ bias, float* __restrict__ out) {
    const int row   = blockIdx.x;
    const int start = rowptr[row];
    const int end   = rowptr[row + 1];
    const int t     = threadIdx.x;

    float4 acc; acc.x = acc.y = acc.z = acc.w = 0.f;
    for (int j = start; j < end; ++j) {
        int   c = cols[j];
        float v = vals[j];
        if (j + 1 < end)  // prefetch next gathered row (global_prefetch_b8)
            __builtin_prefetch(h + (size_t)cols[j + 1] * HID + t * 4, 0, 0);
        float4 hv = ((const float4*)(h + (size_t)c * HID))[t];
        acc.x = fmaf(v, hv.x, acc.x);
        acc.y = fmaf(v, hv.y, acc.y);
        acc.z = fmaf(v, hv.z, acc.z);
        acc.w = fmaf(v, hv.w, acc.w);
    }
    float4 b = ((const float4*)bias)[t];
    acc.x = fmaxf(acc.x + b.x, 0.f);
    acc.y = fmaxf(acc.y + b.y, 0.f);
    acc.z = fmaxf(acc.z + b.z, 0.f);
    acc.w = fmaxf(acc.w + b.w, 0.f);
    ((float4*)(out + (size_t)row * HID))[t] = acc;
}

// ---------------------------------------------------------------------------
// Head: out[r] = sigmoid(dot(x[r,:], w) + b). One wave per row; each lane
// owns a contiguous 16-float chunk (4 x b128 loads), then wave32 xor-reduce.
// ---------------------------------------------------------------------------
__global__ __launch_bounds__(256)
void mv_sigmoid(const float* __restrict__ x, const float* __restrict__ w,
                const float* __restrict__ b, float* __restrict__ out, int M) {
    const int lane = threadIdx.x & 31;
    const int row  = blockIdx.x * 8 + (threadIdx.x >> 5);
    if (row >= M) return;
    const float4* xr4 = (const float4*)(x + (size_t)row * HID);
    const float4* w4  = (const float4*)w;
    float acc = 0.f;
#pragma unroll
    for (int i = 0; i < 4; ++i) {
        float4 xv = xr4[lane * 4 + i];
        float4 wv = w4 [lane * 4 + i];
        acc = fmaf(xv.x, wv.x, acc);
        acc = fmaf(xv.y, wv.y, acc);
        acc = fmaf(xv.z, wv.z, acc);
        acc = fmaf(xv.w, wv.w, acc);
    }
#pragma unroll
    for (int off = 16; off > 0; off >>= 1) acc += __shfl_xor(acc, off, 32);
    if (lane == 0) out[row] = 1.0f / (1.0f + expf(-(acc + b[0])));
}

// ---------------------------------------------------------------------------
extern "C" void kernel_launch(void* const* d_in, const int* in_sizes, int n_in,
                              void* d_out, int out_size, void* d_ws, size_t ws_size,
                              hipStream_t stream) {
    const float* feat    = (const float*)d_in[0];
    const int*   adj_row = (const int*)  d_in[1];
    const int*   adj_col = (const int*)  d_in[2];
    const float* adj_val = (const float*)d_in[3];
    const float* Wg      = (const float*)d_in[4];
    const float* bg      = (const float*)d_in[5];
    const float* vw1     = (const float*)d_in[6];
    const float* vb1     = (const float*)d_in[7];
    const float* vw2     = (const float*)d_in[8];
    const float* vb2     = (const float*)d_in[9];
    float* out = (float*)d_out;

    const int N = in_sizes[0] / HID;            // 10000 nodes
    const int E = in_sizes[1];                  // 160000 edges
    const int L = in_sizes[4] / (HID * HID);    // 5 layers

    // Workspace layout
    char* ws = (char*)d_ws;
    size_t off = 0;
    auto alloc = [&](size_t bytes) -> void* {
        void* p = ws + off;
        off = (off + bytes + 255) & ~(size_t)255;
        return p;
    };
    float* bufH   = (float*)alloc((size_t)N * HID * sizeof(float));
    float* bufX0  = (float*)alloc((size_t)N * HID * sizeof(float));
    float* bufX1  = (float*)alloc((size_t)N * HID * sizeof(float));
    float* wT     = (float*)alloc((size_t)(L + 1) * HID * HID * sizeof(float)); // 5x Wg + vw1
    int*   rowptr = (int*)  alloc((size_t)(N + 1) * sizeof(int));
    int*   counts = (int*)  alloc((size_t)N * sizeof(int));
    int*   fill   = (int*)  alloc((size_t)N * sizeof(int));
    int*   scols  = (int*)  alloc((size_t)E * sizeof(int));
    float* svals  = (float*)alloc((size_t)E * sizeof(float));

    // --- Build CSR once per launch (reused by all 5 layers) ---
    zero_i32<<<(N + 255) / 256, 256, 0, stream>>>(counts, N);
    zero_i32<<<(N + 255) / 256, 256, 0, stream>>>(fill, N);
    count_edges<<<(E + 255) / 256, 256, 0, stream>>>(adj_row, counts, E);
    scan_rows<<<1, 1024, 0, stream>>>(counts, rowptr, N);
    scatter_edges<<<(E + 255) / 256, 256, 0, stream>>>(adj_row, adj_col, adj_val,
                                                       rowptr, fill, scols, svals, E);

    // --- Transpose all weights once (K-contiguous B fragments for WMMA) ---
    {
        dim3 tg(HID / 32, HID / 32);
        for (int i = 0; i < L; ++i)
            transpose512<<<tg, 256, 0, stream>>>(Wg + (size_t)i * HID * HID,
                                                 wT + (size_t)i * HID * HID);
        transpose512<<<tg, 256, 0, stream>>>(vw1, wT + (size_t)L * HID * HID);
    }

    // --- 5 GraphConv layers: GEMM -> SpMM(+bias,ReLU) ---
    const int gemmGrid = (N + 15) / 16;
    const float* xcur = feat;
    float* xnext = bufX0;
    for (int i = 0; i < L; ++i) {
        gemm_wmma_f32<<<gemmGrid, 256, 0, stream>>>(
            xcur, wT + (size_t)i * HID * HID, nullptr, bufH, N, 0);
        spmm_bias_relu<<<N, 128, 0, stream>>>(
            rowptr, scols, svals, bufH, bg + (size_t)i * HID, xnext);
        xcur  = xnext;
        xnext = (xnext == bufX0) ? bufX1 : bufX0;
    }

    // --- Head: relu(x @ vw1 + vb1) then sigmoid(h @ vw2 + vb2) ---
    gemm_wmma_f32<<<gemmGrid, 256, 0, stream>>>(
        xcur, wT + (size_t)L * HID * HID, vb1, bufH, N, 1);
    mv_sigmoid<<<(N + 7) / 8, 256, 0, stream>>>(bufH, vw2, vb2, out, N);
}